// PhysicsTransitionPrior_33689723470432
// MI455X (gfx1250) — compile-verified
//
#include <hip/hip_runtime.h>
#include <math.h>

#define BATCH 512
#define NPART 64
#define FLU   12288          // 16^3 * 3
#define OUT_STRIDE 13313     // 192+192+256+192+192+1+12288
#define DT_   (1.0f/60.0f)
#define G_ACC_ 9.81f
#define RHO_  1.225f
#define CD_   0.47f
#define NU_   0.001f

typedef __attribute__((ext_vector_type(2))) float v2f;
typedef __attribute__((ext_vector_type(8))) float v8f;

__device__ __forceinline__ float softplusf(float x) {
  return (x > 20.f) ? x : log1pf(expf(x));
}

// ---------------------------------------------------------------------------
// Rigid-body RK4: one block per batch, 128 threads (4 waves).
// Contact damping matmul D = C(64x64) x [vel | 1 | 0...](64x16) via
// v_wmma_f32_16x16x4_f32 (exact fp32). Column 3 of ones yields rowsum(C).
// sV is padded to all 16 WMMA columns so B loads are unconditional (no EXEC
// juggling around the wmma loop).
// ---------------------------------------------------------------------------
__global__ __launch_bounds__(128) void phys_particles(
    const float* __restrict__ g_pos, const float* __restrict__ g_vel,
    const float* __restrict__ g_rot, const float* __restrict__ g_omega,
    const float* __restrict__ g_mass, const float* __restrict__ g_inertia,
    const float* __restrict__ g_contacts, const float* __restrict__ g_energy,
    const float* __restrict__ g_Fext, const float* __restrict__ g_tau,
    const float* __restrict__ g_logA, const float* __restrict__ g_logk,
    const float* __restrict__ g_logb, float* __restrict__ g_out)
{
  const int b = blockIdx.x;
  const int t = threadIdx.x;
  const int i = t & 63;
  const int half = t >> 6;

  __shared__ float sC[64][64];   // contacts row-major (WMMA A source)
  __shared__ float sP[64][3];    // current-stage positions
  __shared__ float sV[64][16];   // B panel: [vx vy vz 1 0 ... 0] per row
  __shared__ float sD[64][4];    // matmul result: C*v (cols 0..2), rowsum (col 3)
  __shared__ float sSp[64][3];   // spring partial sums (half==1)
  __shared__ float sE[64];       // energy reduction

  const float* Cb = g_contacts + (size_t)b * 4096;
  for (int idx = t; idx < 4096; idx += 128)
    sC[idx >> 6][idx & 63] = Cb[idx];
  // one-time fill of constant B columns (3..15): col3 = 1, rest = 0
  for (int idx = t; idx < 1024; idx += 128) {
    int col = idx & 15;
    sV[idx >> 4][col] = (col == 3) ? 1.0f : 0.0f;
  }

  const float kk = softplusf(g_logk[0]);
  const float bb = softplusf(g_logb[0]);

  // owner-thread (t<64) registers
  float p0[3], v0[3], r0[4], w0[3], Iv[3], Fx[3], tq[3];
  float m = 1.f, Ai = 0.f;
  float cp[3], cv[3], cr[4], cw[3];
  float aP[3] = {0,0,0}, aV[3] = {0,0,0}, aR[4] = {0,0,0,0}, aW[3] = {0,0,0};
  float k1dv[3] = {0,0,0};
  float accE = 0.f;

  if (t < 64) {
    size_t b3 = ((size_t)b * 64 + i) * 3;
    size_t b4 = ((size_t)b * 64 + i) * 4;
    for (int n = 0; n < 3; ++n) {
      p0[n] = g_pos[b3+n];  v0[n] = g_vel[b3+n];  w0[n] = g_omega[b3+n];
      Iv[n] = g_inertia[b3+n]; Fx[n] = g_Fext[b3+n]; tq[n] = g_tau[b3+n];
    }
    for (int c = 0; c < 4; ++c) r0[c] = g_rot[b4+c];
    m  = g_mass[(size_t)b * 64 + i];
    Ai = softplusf(g_logA[i]);
    for (int n = 0; n < 3; ++n) { cp[n] = p0[n]; cv[n] = v0[n]; cw[n] = w0[n]; }
    for (int c = 0; c < 4; ++c) cr[c] = r0[c];
  }
  __syncthreads();   // sV constant columns complete before owners write 0..2

  const int wave = t >> 5;
  const int lane = t & 31;
  const int hi   = lane >> 4;
  const int mn   = lane & 15;

  for (int s = 0; s < 4; ++s) {
    if (t < 64) {
      sP[i][0] = cp[0]; sP[i][1] = cp[1]; sP[i][2] = cp[2];
      sV[i][0] = cv[0]; sV[i][1] = cv[1]; sV[i][2] = cv[2];
    }
    __syncthreads();

    // ---- WMMA: each wave computes a 16x16 tile of D = C * V1 ----
    v8f acc = {};
    #pragma unroll
    for (int k0 = 0; k0 < 64; k0 += 4) {
      const int kr = k0 + 2 * hi;
      v2f a, bm;
      a.x = sC[wave * 16 + mn][kr];     // A[m][k], m = lane&15, k split per half
      a.y = sC[wave * 16 + mn][kr + 1];
      bm.x = sV[kr][mn];                // B[k][n], n = lane&15 (cols>3 are zero)
      bm.y = sV[kr + 1][mn];
      acc = __builtin_amdgcn_wmma_f32_16x16x4_f32(
          false, a, false, bm, (short)0, acc, false, false);
    }
    if (mn < 4) {
      #pragma unroll
      for (int r = 0; r < 8; ++r)
        sD[wave * 16 + r + 8 * hi][mn] = acc[r];   // D: M = r + 8*hi, N = mn
    }

    // ---- spring term: 2 threads per particle, 32 j's each ----
    float px = sP[i][0], py = sP[i][1], pz = sP[i][2];
    float sx = 0.f, sy = 0.f, sz = 0.f;
    const int j0 = 32 * half;
    for (int j = j0; j < j0 + 32; ++j) {
      float dx = sP[j][0] - px, dy = sP[j][1] - py, dz = sP[j][2] - pz;
      float dist = sqrtf(dx*dx + dy*dy + dz*dz);
      dist = fmaxf(dist, 1e-6f);
      float pen = fmaxf(1.f - dist, 0.f);
      float wgt = kk * pen / dist;
      sx += wgt * dx; sy += wgt * dy; sz += wgt * dz;
    }
    if (half) { sSp[i][0] = sx; sSp[i][1] = sy; sSp[i][2] = sz; }
    __syncthreads();

    const float wrk = (s == 1 || s == 2) ? 2.f : 1.f;
    if (t < 64) {
      float rowsum = sD[i][3];
      float Fc[3], Fd[3], dv[3], dw[3], dr[4];
      Fc[0] = (sx + sSp[i][0]) + bb * (sD[i][0] - cv[0] * rowsum);
      Fc[1] = (sy + sSp[i][1]) + bb * (sD[i][1] - cv[1] * rowsum);
      Fc[2] = (sz + sSp[i][2]) + bb * (sD[i][2] - cv[2] * rowsum);
      float vm = fmaxf(sqrtf(cv[0]*cv[0] + cv[1]*cv[1] + cv[2]*cv[2]), 1e-6f);
      float dc = -0.5f * RHO_ * CD_ * Ai * vm;
      Fd[0] = dc*cv[0]; Fd[1] = dc*cv[1]; Fd[2] = dc*cv[2];
      float invm = 1.f / m;
      dv[0] = (Fx[0] + Fd[0] + Fc[0]) * invm;
      dv[1] = (Fx[1] + Fd[1] + Fc[1]) * invm;
      dv[2] = (Fx[2] - m * G_ACC_ + Fd[2] + Fc[2]) * invm;
      float Iox = Iv[0]*cw[0], Ioy = Iv[1]*cw[1], Ioz = Iv[2]*cw[2];
      dw[0] = (tq[0] - (cw[1]*Ioz - cw[2]*Ioy)) / fmaxf(Iv[0], 1e-6f);
      dw[1] = (tq[1] - (cw[2]*Iox - cw[0]*Ioz)) / fmaxf(Iv[1], 1e-6f);
      dw[2] = (tq[2] - (cw[0]*Ioy - cw[1]*Iox)) / fmaxf(Iv[2], 1e-6f);
      float qw = cr[0], qx = cr[1], qy = cr[2], qz = cr[3];
      dr[0] = 0.5f * (-qx*cw[0] - qy*cw[1] - qz*cw[2]);
      dr[1] = 0.5f * ( qw*cw[0] + qy*cw[2] - qz*cw[1]);
      dr[2] = 0.5f * ( qw*cw[1] - qx*cw[2] + qz*cw[0]);
      dr[3] = 0.5f * ( qw*cw[2] + qx*cw[1] - qy*cw[0]);
      sE[i] = (Fx[0]+Fd[0])*cv[0] + (Fx[1]+Fd[1])*cv[1] + (Fx[2]+Fd[2])*cv[2];
      if (s == 0) { k1dv[0] = dv[0]; k1dv[1] = dv[1]; k1dv[2] = dv[2]; }
      for (int n = 0; n < 3; ++n) { aP[n] += wrk*cv[n]; aV[n] += wrk*dv[n]; aW[n] += wrk*dw[n]; }
      for (int c = 0; c < 4; ++c) aR[c] += wrk*dr[c];
      if (s < 3) {   // _apply always restarts from the ORIGINAL state
        float a_s = (s < 2) ? (0.5f * DT_) : DT_;
        float np0 = p0[0] + a_s*cv[0], np1 = p0[1] + a_s*cv[1], np2 = p0[2] + a_s*cv[2];
        cv[0] = v0[0] + a_s*dv[0]; cv[1] = v0[1] + a_s*dv[1]; cv[2] = v0[2] + a_s*dv[2];
        cp[0] = np0; cp[1] = np1; cp[2] = np2;
        cw[0] = w0[0] + a_s*dw[0]; cw[1] = w0[1] + a_s*dw[1]; cw[2] = w0[2] + a_s*dw[2];
        float q0 = r0[0] + a_s*dr[0], q1 = r0[1] + a_s*dr[1];
        float q2 = r0[2] + a_s*dr[2], q3 = r0[3] + a_s*dr[3];
        float qn = fmaxf(sqrtf(q0*q0 + q1*q1 + q2*q2 + q3*q3), 1e-12f);
        cr[0] = q0/qn; cr[1] = q1/qn; cr[2] = q2/qn; cr[3] = q3/qn;
      }
    }
    __syncthreads();
    for (int off = 32; off >= 1; off >>= 1) {
      if (t < off) sE[t] += sE[t + off];
      __syncthreads();
    }
    if (t == 0) accE += wrk * sE[0];
    __syncthreads();
  }

  if (t < 64) {
    float* ob = g_out + (size_t)b * OUT_STRIDE;
    const float c6 = DT_ / 6.f;
    for (int n = 0; n < 3; ++n) ob[        i*3 + n] = p0[n] + c6*aP[n];
    for (int n = 0; n < 3; ++n) ob[192 +   i*3 + n] = v0[n] + c6*aV[n];
    float q0 = r0[0] + c6*aR[0], q1 = r0[1] + c6*aR[1];
    float q2 = r0[2] + c6*aR[2], q3 = r0[3] + c6*aR[3];
    float qn = fmaxf(sqrtf(q0*q0 + q1*q1 + q2*q2 + q3*q3), 1e-12f);
    ob[384 + i*4 + 0] = q0/qn; ob[384 + i*4 + 1] = q1/qn;
    ob[384 + i*4 + 2] = q2/qn; ob[384 + i*4 + 3] = q3/qn;
    for (int n = 0; n < 3; ++n) ob[640 + i*3 + n] = w0[n] + c6*aW[n];
    for (int n = 0; n < 3; ++n) ob[832 + i*3 + n] = Fx[n] + m*k1dv[n];
    if (t == 0) ob[1024] = g_energy[b] + c6*accE;
  }
}

// ---------------------------------------------------------------------------
// Fluid: depthwise 3x3x3 conv, u resident in LDS (48KB), one block per batch.
// ---------------------------------------------------------------------------
__global__ __launch_bounds__(256) void fluid_deriv(
    const float* __restrict__ u, const float* __restrict__ cwts,
    float* __restrict__ kout)
{
  const int b = blockIdx.x, t = threadIdx.x;
  __shared__ float su[3][4096];
  __shared__ float sw[3][27];
  const float* ub = u + (size_t)b * FLU;
  for (int idx = t; idx < FLU; idx += 256) {
    int cell = idx / 3;
    su[idx - cell*3][cell] = ub[idx];
  }
  if (t < 81) sw[t / 27][t % 27] = cwts[t];
  __syncthreads();
  float* kb = kout + (size_t)b * FLU;
  for (int p = t; p < 4096; p += 256) {
    const int d = p >> 8, h = (p >> 4) & 15, w = p & 15;
    float a0 = 0.f, a1 = 0.f, a2 = 0.f;
    #pragma unroll
    for (int kd = 0; kd < 3; ++kd) {
      int dd = d + kd - 1;
      if ((unsigned)dd > 15u) continue;
      #pragma unroll
      for (int kh = 0; kh < 3; ++kh) {
        int hh = h + kh - 1;
        if ((unsigned)hh > 15u) continue;
        #pragma unroll
        for (int kw = 0; kw < 3; ++kw) {
          int ww = w + kw - 1;
          if ((unsigned)ww > 15u) continue;
          int cell = (dd << 8) | (hh << 4) | ww;
          int wi = kd*9 + kh*3 + kw;
          a0 += sw[0][wi] * su[0][cell];
          a1 += sw[1][wi] * su[1][cell];
          a2 += sw[2][wi] * su[2][cell];
        }
      }
    }
    kb[p*3 + 0] = NU_*a0; kb[p*3 + 1] = NU_*a1; kb[p*3 + 2] = NU_*a2;
  }
}

// ---------------------------------------------------------------------------
// RK4 accumulate (acc lives in d_out fluid slots), next-state + de-mean.
// isFinal: acc += k4; out = demean(u0 + DT/6*acc) written over acc slots.
// ---------------------------------------------------------------------------
__global__ __launch_bounds__(256) void fluid_apply(
    const float* __restrict__ u0, const float* __restrict__ kb,
    float* __restrict__ out, float* __restrict__ unext,
    float a_s, float wrk, int initAcc, int isFinal)
{
  const int b = blockIdx.x, t = threadIdx.x;
  __shared__ float ssum[256][3];
  const float* u0b = u0 + (size_t)b * FLU;
  const float* kbb = kb + (size_t)b * FLU;
  float* accp = out + (size_t)b * OUT_STRIDE + 1025;
  float* unb  = unext + (size_t)b * FLU;
  float raw[48];                        // FLU/256 == 48 exactly
  float s0 = 0.f, s1 = 0.f, s2 = 0.f;
  int cnt = 0;
  for (int idx = t; idx < FLU; idx += 256) {
    float kv = kbb[idx];
    float an = (initAcc ? 0.f : accp[idx]) + wrk * kv;
    float r;
    if (isFinal) {
      r = u0b[idx] + (DT_ / 6.f) * an;
    } else {
      accp[idx] = an;
      r = u0b[idx] + a_s * kv;
    }
    raw[cnt++] = r;
    int c = idx % 3;
    if (c == 0) s0 += r; else if (c == 1) s1 += r; else s2 += r;
  }
  ssum[t][0] = s0; ssum[t][1] = s1; ssum[t][2] = s2;
  __syncthreads();
  for (int off = 128; off >= 1; off >>= 1) {
    if (t < off) {
      ssum[t][0] += ssum[t+off][0];
      ssum[t][1] += ssum[t+off][1];
      ssum[t][2] += ssum[t+off][2];
    }
    __syncthreads();
  }
  const float m0 = ssum[0][0] * (1.f/4096.f);
  const float m1 = ssum[0][1] * (1.f/4096.f);
  const float m2 = ssum[0][2] * (1.f/4096.f);
  cnt = 0;
  for (int idx = t; idx < FLU; idx += 256) {
    int c = idx % 3;
    float r = raw[cnt++] - (c == 0 ? m0 : (c == 1 ? m1 : m2));
    if (isFinal) accp[idx] = r; else unb[idx] = r;
  }
}

extern "C" void kernel_launch(void* const* d_in, const int* in_sizes, int n_in,
                              void* d_out, int out_size, void* d_ws, size_t ws_size,
                              hipStream_t stream) {
  (void)in_sizes; (void)n_in; (void)out_size; (void)ws_size;
  const float* pos      = (const float*)d_in[0];
  const float* vel      = (const float*)d_in[1];
  const float* rot      = (const float*)d_in[2];
  const float* omega    = (const float*)d_in[3];
  const float* mass     = (const float*)d_in[4];
  const float* inertia  = (const float*)d_in[5];
  const float* contacts = (const float*)d_in[6];
  const float* energy   = (const float*)d_in[7];
  const float* fluid    = (const float*)d_in[8];
  const float* Fext     = (const float*)d_in[9];
  const float* tau      = (const float*)d_in[10];
  const float* logA     = (const float*)d_in[11];
  const float* logk     = (const float*)d_in[12];
  const float* logb     = (const float*)d_in[13];
  const float* convw    = (const float*)d_in[14];
  float* out   = (float*)d_out;
  float* u_cur = (float*)d_ws;                       // BATCH*FLU floats
  float* kbuf  = u_cur + (size_t)BATCH * FLU;        // BATCH*FLU floats

  phys_particles<<<BATCH, 128, 0, stream>>>(pos, vel, rot, omega, mass, inertia,
      contacts, energy, Fext, tau, logA, logk, logb, out);

  // k1
  fluid_deriv<<<BATCH, 256, 0, stream>>>(fluid, convw, kbuf);
  fluid_apply<<<BATCH, 256, 0, stream>>>(fluid, kbuf, out, u_cur, 0.5f*DT_, 1.f, 1, 0);
  // k2
  fluid_deriv<<<BATCH, 256, 0, stream>>>(u_cur, convw, kbuf);
  fluid_apply<<<BATCH, 256, 0, stream>>>(fluid, kbuf, out, u_cur, 0.5f*DT_, 2.f, 0, 0);
  // k3
  fluid_deriv<<<BATCH, 256, 0, stream>>>(u_cur, convw, kbuf);
  fluid_apply<<<BATCH, 256, 0, stream>>>(fluid, kbuf, out, u_cur, DT_, 2.f, 0, 0);
  // k4 + final combine (writes fluid output)
  fluid_deriv<<<BATCH, 256, 0, stream>>>(u_cur, convw, kbuf);
  fluid_apply<<<BATCH, 256, 0, stream>>>(fluid, kbuf, out, u_cur, 0.f, 1.f, 0, 1);
}